// MaskedDeepRNN_2911987826990
// MI455X (gfx1250) — compile-verified
//
#include <hip/hip_runtime.h>
#include <math.h>

// MaskedDeepRNN on gfx1250 (MI455X), fp32 end-to-end via V_WMMA_F32_16X16X4_F32.
//
// Pipeline (all on `stream`, graph-capture safe, deterministic):
//   0) mask weights:  wm = w * mask                       (4 tiny kernels)
//   1) igate0 = x · wm_ih0^T + b_ih0        -> ws.ig      (big WMMA GEMM)
//   2) 512x:  h = tanh(ig0[t] + h·wm_hh0^T + b_hh0)       -> ws.hall[t]
//   3) igate1 = hall · wm_ih1^T + b_ih1     -> ws.ig      (reuse, WMMA GEMM)
//   4) 512x:  h = tanh(ig1[t] + h·wm_hh1^T + b_hh1)       -> ws.hall[t] (reuse)
//              t==511 also writes d_out [B,H]
//
// Workspace needs: 2 * 128MB (ig, hall) + ~14MB masked weights.

typedef float v2f __attribute__((ext_vector_type(2)));
typedef float v8f __attribute__((ext_vector_type(8)));

static constexpr int BB = 64;    // batch
static constexpr int TT = 512;   // timesteps
static constexpr int DD = 512;   // input dim
static constexpr int HH = 1024;  // hidden dim

#define WMMA4(ACC, A_, B_)                                                     \
  (ACC) = __builtin_amdgcn_wmma_f32_16x16x4_f32(false, (A_), false, (B_),      \
                                                (short)0, (ACC), false, false)

// ---------------------------------------------------------------------------
// wm[i] = mask[i] ? w[i] : 0
__global__ __launch_bounds__(256)
void mask_weights_kernel(const float* __restrict__ w,
                         const unsigned char* __restrict__ m,
                         float* __restrict__ o, int n) {
  const int i = blockIdx.x * 256 + threadIdx.x;
  if (i < n) o[i] = m[i] ? w[i] : 0.0f;
}

// ---------------------------------------------------------------------------
// out[m, n] = sum_k A[t*sT + b*sB + k] * W[n*K + k] + bias[n]
// with m = t*64 + b, m in [0, T*B), n in [0, 1024).
// One wave -> 16(M) x 64(N) tile, K stepped by 4 (f32 WMMA), 8 waves/WG.
__global__ __launch_bounds__(256)
void igate_gemm_wmma(const float* __restrict__ A, const float* __restrict__ W,
                     const float* __restrict__ bias, float* __restrict__ out,
                     int K, int sT, int sB) {
  const int lane = threadIdx.x & 31;
  const int wv   = threadIdx.x >> 5;
  const int gw   = blockIdx.x * 8 + wv;   // global wave id
  const int mblk = gw >> 4;               // 16 N-stripes per M-block
  const int nstr = gw & 15;

  const int r  = lane & 15;               // row (A) / col (B) within fragment
  const int hi = lane >> 4;               // lane half selects K pair
  const int kk = hi << 1;

  const int m0   = mblk << 4;             // 16 consecutive m share t (16 | 64)
  const int t    = m0 >> 6;
  const int brow = (m0 + r) & 63;

  const float* Arow = A + (size_t)t * (size_t)sT + (size_t)brow * (size_t)sB + kk;
  const int nb = nstr << 6;
  const float* W0 = W + (size_t)(nb +  0 + r) * (size_t)K + kk;
  const float* W1 = W + (size_t)(nb + 16 + r) * (size_t)K + kk;
  const float* W2 = W + (size_t)(nb + 32 + r) * (size_t)K + kk;
  const float* W3 = W + (size_t)(nb + 48 + r) * (size_t)K + kk;

  v8f acc0 = {}, acc1 = {}, acc2 = {}, acc3 = {};

  for (int k0 = 0; k0 < K; k0 += 4) {
    const v2f a  = *(const v2f*)(Arow + k0);
    const v2f b0 = *(const v2f*)(W0 + k0);
    const v2f b1 = *(const v2f*)(W1 + k0);
    const v2f b2 = *(const v2f*)(W2 + k0);
    const v2f b3 = *(const v2f*)(W3 + k0);
    WMMA4(acc0, a, b0);
    WMMA4(acc1, a, b1);
    WMMA4(acc2, a, b2);
    WMMA4(acc3, a, b3);
  }

  const float bs0 = bias[nb +  0 + r];
  const float bs1 = bias[nb + 16 + r];
  const float bs2 = bias[nb + 32 + r];
  const float bs3 = bias[nb + 48 + r];

#pragma unroll
  for (int rr = 0; rr < 8; ++rr) {
    const int m = m0 + rr + (hi << 3);          // C/D layout: M = rr + 8*hi
    float* orow = out + (size_t)m * HH + nb + r;
    orow[0]  = acc0[rr] + bs0;
    orow[16] = acc1[rr] + bs1;
    orow[32] = acc2[rr] + bs2;
    orow[48] = acc3[rr] + bs3;
  }
}

// ---------------------------------------------------------------------------
// One recurrence step:  hout[b, n] = tanh(ig[b, n] + sum_k hprev[b,k]*Whh[n,k] + bias[n])
// M=64, N=K=1024. Grid: 4 M-blocks x 4 N-quads; 4 waves/WG, one N-stripe (64) each.
// hprev 16x1024 slice staged transposed in LDS (lds[k*16+row]) -> conflict-free
// ds reads of the A fragment. first!=0 => hprev treated as zeros.
__global__ __launch_bounds__(128)
void rnn_step_wmma(const float* __restrict__ hprev, const float* __restrict__ Whh,
                   const float* __restrict__ bias, const float* __restrict__ ig,
                   float* __restrict__ hout, float* __restrict__ out2, int first) {
  __shared__ float lds_a[16 * HH];   // 64 KB, transposed: lds_a[k*16 + row]

  const int tid  = threadIdx.x;
  const int lane = tid & 31;
  const int wv   = tid >> 5;
  const int mblk = blockIdx.x >> 2;
  const int nstr = ((blockIdx.x & 3) << 2) + wv;

  const int r  = lane & 15;
  const int hi = lane >> 4;
  const int kk = hi << 1;
  const int nb = nstr << 6;

  v8f acc0 = {}, acc1 = {}, acc2 = {}, acc3 = {};

  if (!first) {
    // cooperative transposed stage of hprev rows [mblk*16, mblk*16+16)
#pragma unroll
    for (int i = 0; i < 32; ++i) {
      const int e   = i * 512 + tid * 4;   // bijective over 4096 float4s
      const int row = e >> 10;
      const int col = e & 1023;
      const float4 v = *(const float4*)(hprev + (size_t)(mblk * 16 + row) * HH + col);
      lds_a[(col + 0) * 16 + row] = v.x;
      lds_a[(col + 1) * 16 + row] = v.y;
      lds_a[(col + 2) * 16 + row] = v.z;
      lds_a[(col + 3) * 16 + row] = v.w;
    }
    __syncthreads();

    const float* W0 = Whh + (size_t)(nb +  0 + r) * HH + kk;
    const float* W1 = Whh + (size_t)(nb + 16 + r) * HH + kk;
    const float* W2 = Whh + (size_t)(nb + 32 + r) * HH + kk;
    const float* W3 = Whh + (size_t)(nb + 48 + r) * HH + kk;

    for (int k0 = 0; k0 < HH; k0 += 4) {
      v2f a;
      a.x = lds_a[(k0 + kk + 0) * 16 + r];
      a.y = lds_a[(k0 + kk + 1) * 16 + r];
      const v2f b0 = *(const v2f*)(W0 + k0);
      const v2f b1 = *(const v2f*)(W1 + k0);
      const v2f b2 = *(const v2f*)(W2 + k0);
      const v2f b3 = *(const v2f*)(W3 + k0);
      WMMA4(acc0, a, b0);
      WMMA4(acc1, a, b1);
      WMMA4(acc2, a, b2);
      WMMA4(acc3, a, b3);
    }
  }

  const float bs0 = bias[nb +  0 + r];
  const float bs1 = bias[nb + 16 + r];
  const float bs2 = bias[nb + 32 + r];
  const float bs3 = bias[nb + 48 + r];

#pragma unroll
  for (int rr = 0; rr < 8; ++rr) {
    const int brow = mblk * 16 + rr + (hi << 3);
    const float* igr = ig + (size_t)brow * HH + nb + r;
    float* orow = hout + (size_t)brow * HH + nb + r;
    const float o0 = tanhf(acc0[rr] + igr[0]  + bs0);
    const float o1 = tanhf(acc1[rr] + igr[16] + bs1);
    const float o2 = tanhf(acc2[rr] + igr[32] + bs2);
    const float o3 = tanhf(acc3[rr] + igr[48] + bs3);
    orow[0] = o0; orow[16] = o1; orow[32] = o2; orow[48] = o3;
    if (out2) {
      float* xrow = out2 + (size_t)brow * HH + nb + r;
      xrow[0] = o0; xrow[16] = o1; xrow[32] = o2; xrow[48] = o3;
    }
  }
}

// ---------------------------------------------------------------------------
extern "C" void kernel_launch(void* const* d_in, const int* in_sizes, int n_in,
                              void* d_out, int out_size, void* d_ws, size_t ws_size,
                              hipStream_t stream) {
  (void)in_sizes; (void)n_in; (void)out_size; (void)ws_size;

  const float* x     = (const float*)d_in[0];
  const float* w_ih0 = (const float*)d_in[1];
  const float* w_hh0 = (const float*)d_in[2];
  const float* b_ih0 = (const float*)d_in[3];
  const float* b_hh0 = (const float*)d_in[4];
  const float* w_ih1 = (const float*)d_in[5];
  const float* w_hh1 = (const float*)d_in[6];
  const float* b_ih1 = (const float*)d_in[7];
  const float* b_hh1 = (const float*)d_in[8];
  const unsigned char* m_i0 = (const unsigned char*)d_in[9];
  const unsigned char* m_h0 = (const unsigned char*)d_in[10];
  const unsigned char* m_i1 = (const unsigned char*)d_in[11];
  const unsigned char* m_h1 = (const unsigned char*)d_in[12];

  const size_t igElems = (size_t)TT * BB * HH;           // 33,554,432 floats
  char* ws = (char*)d_ws;
  float* ig   = (float*)(ws);                            // 128 MB
  float* hall = (float*)(ws + igElems * sizeof(float));  // 128 MB
  float* wmi0 = (float*)(ws + 2 * igElems * sizeof(float));
  float* wmh0 = wmi0 + (size_t)HH * DD;
  float* wmi1 = wmh0 + (size_t)HH * HH;
  float* wmh1 = wmi1 + (size_t)HH * HH;

  // 0) apply sparsity masks
  mask_weights_kernel<<<(HH * DD + 255) / 256, 256, 0, stream>>>(w_ih0, m_i0, wmi0, HH * DD);
  mask_weights_kernel<<<(HH * HH + 255) / 256, 256, 0, stream>>>(w_hh0, m_h0, wmh0, HH * HH);
  mask_weights_kernel<<<(HH * HH + 255) / 256, 256, 0, stream>>>(w_ih1, m_i1, wmi1, HH * HH);
  mask_weights_kernel<<<(HH * HH + 255) / 256, 256, 0, stream>>>(w_hh1, m_h1, wmh1, HH * HH);

  // waves = (T*B/16) M-blocks * 16 N-stripes; 8 waves per 256-thread WG
  const int nBlocksGemm = (TT * BB / 16) * 16 / 8;       // 4096

  // 1) layer-0 input gates: A row (t,b) = x[b, t, :]  (x is [B, T, D])
  igate_gemm_wmma<<<nBlocksGemm, 256, 0, stream>>>(x, wmi0, b_ih0, ig,
                                                   DD, /*sT=*/DD, /*sB=*/TT * DD);

  // 2) layer-0 scan
  for (int t = 0; t < TT; ++t) {
    const float* hp = t ? hall + (size_t)(t - 1) * BB * HH : nullptr;
    rnn_step_wmma<<<16, 128, 0, stream>>>(hp, wmh0, b_hh0,
                                          ig + (size_t)t * BB * HH,
                                          hall + (size_t)t * BB * HH,
                                          nullptr, t == 0);
  }

  // 3) layer-1 input gates: A = hall, row m = t*64+b contiguous at m*H
  igate_gemm_wmma<<<nBlocksGemm, 256, 0, stream>>>(hall, wmi1, b_ih1, ig,
                                                   HH, /*sT=*/BB * HH, /*sB=*/HH);

  // 4) layer-1 scan (overwrites hall in place; h1_all already consumed)
  for (int t = 0; t < TT; ++t) {
    const float* hp = t ? hall + (size_t)(t - 1) * BB * HH : nullptr;
    float* o2 = (t == TT - 1) ? (float*)d_out : nullptr;
    rnn_step_wmma<<<16, 128, 0, stream>>>(hp, wmh1, b_hh1,
                                          ig + (size_t)t * BB * HH,
                                          hall + (size_t)t * BB * HH,
                                          o2, t == 0);
  }
}